// GraphAttentionLayer_79508434584007
// MI455X (gfx1250) — compile-verified
//
#include <hip/hip_runtime.h>

typedef __attribute__((ext_vector_type(2))) float v2f;
typedef __attribute__((ext_vector_type(8))) float v8f;

// D(16x16,f32) = A(16x4,f32) x B(4x16,f32) + C   -- CDNA5 V_WMMA_F32_16X16X4_F32
#define WMMA_F32_4(a, bmat, c) \
  __builtin_amdgcn_wmma_f32_16x16x4_f32(false, (a), false, (bmat), (short)0, (c), false, false)

constexpr int   kB    = 8;
constexpr int   kN    = 1024;
constexpr int   kInF  = 768;
constexpr int   kOutF = 64;
constexpr float kAlpha  = 0.2f;
constexpr float kNegInf = -9.0e15f;

// ---------------------------------------------------------------------------
// Kernel 1: H = X * W^T + bias      (8192x768 @ 768x64 -> 8192x64)
// One wave per 32-row tile: two 16-row A tiles share each set of B loads
// (8 WMMAs per 6 loads).  A layout (16x4 f32): lane<16 -> M=lane,
// K={k0,k0+1}; lane>=16 -> K={k0+2,k0+3}.  B layout (4x16 f32): lane&15 -> N,
// same K-half split -> contiguous float2 in row-major W (o,i).
// ---------------------------------------------------------------------------
__global__ __launch_bounds__(256) void k1_gemm_h(
    const float* __restrict__ X, const float* __restrict__ W,
    const float* __restrict__ Wb, float* __restrict__ H)
{
  const int lane = threadIdx.x & 31;
  const int wave = threadIdx.x >> 5;
  const int tile = blockIdx.x * 8 + wave;          // 0..255
  const int row0 = tile << 5;                      // 32 rows per wave
  const int half = lane >> 4;                      // K half select
  const int ncol = lane & 15;                      // N col for B, M row for A
  const float* __restrict__ Arow0 = X + (size_t)(row0 + ncol) * kInF;
  const float* __restrict__ Arow1 = Arow0 + (size_t)16 * kInF;

  v8f acc0[4] = {};
  v8f acc1[4] = {};
  for (int k0 = 0; k0 < kInF; k0 += 4) {
    const int kb = k0 + half * 2;
    const v2f a0 = *(const v2f*)(Arow0 + kb);
    const v2f a1 = *(const v2f*)(Arow1 + kb);
#pragma unroll
    for (int t = 0; t < 4; ++t) {
      const v2f bm = *(const v2f*)(W + (size_t)(t * 16 + ncol) * kInF + kb);
      acc0[t] = WMMA_F32_4(a0, bm, acc0[t]);
      acc1[t] = WMMA_F32_4(a1, bm, acc1[t]);
    }
  }
  // C/D layout: VGPR r, lanes 0-15 -> M=r, lanes 16-31 -> M=r+8
  const int rbase = row0 + half * 8;
#pragma unroll
  for (int t = 0; t < 4; ++t) {
    const float bias = Wb[t * 16 + ncol];
#pragma unroll
    for (int r = 0; r < 8; ++r) {
      H[(size_t)(rbase + r) * kOutF + t * 16 + ncol]      = acc0[t][r] + bias;
      H[(size_t)(rbase + 16 + r) * kOutF + t * 16 + ncol] = acc1[t][r] + bias;
    }
  }
}

// ---------------------------------------------------------------------------
// Kernel 2: f1 = H.a1, f2 = H.a2  (per row)
// ---------------------------------------------------------------------------
__global__ __launch_bounds__(256) void k2_f12(
    const float* __restrict__ H, const float* __restrict__ aw,
    float* __restrict__ F1, float* __restrict__ F2)
{
  const int row = blockIdx.x * 256 + threadIdx.x;  // 0..8191
  const float* __restrict__ hr = H + (size_t)row * kOutF;
  float s1 = 0.f, s2 = 0.f;
#pragma unroll 8
  for (int o = 0; o < kOutF; ++o) {
    const float v = hr[o];
    s1 += v * aw[o];
    s2 += v * aw[kOutF + o];
  }
  F1[row] = s1;
  F2[row] = s2;
}

// ---------------------------------------------------------------------------
// Kernel 3: per-row softmax stats (max m, sum l): online lane-local
// reduction + wave32 shfl_xor combine. One wave per row; adj read coalesced.
// ---------------------------------------------------------------------------
__global__ __launch_bounds__(256) void k3_stats(
    const float* __restrict__ F1, const float* __restrict__ F2,
    const int* __restrict__ adj, const float* __restrict__ abp,
    float* __restrict__ Mrow, float* __restrict__ Lrow)
{
  const int lane = threadIdx.x & 31;
  const int wave = threadIdx.x >> 5;
  const int row  = blockIdx.x * 8 + wave;          // 0..8191
  const int b    = row >> 10;
  const float f1i = F1[row] + abp[0];
  const int*   __restrict__ ar  = adj + (size_t)row * kN;
  const float* __restrict__ f2b = F2 + (b << 10);

  float m = -3.0e38f, s = 0.f;
  for (int j = lane; j < kN; j += 32) {
    float e = f1i + f2b[j];
    e = (e >= 0.f) ? e : kAlpha * e;               // leaky_relu
    e = (ar[j] > 0) ? e : kNegInf;                 // adjacency mask
    if (e > m) { s = s * __expf(m - e) + 1.f; m = e; }
    else       { s += __expf(e - m); }
  }
#pragma unroll
  for (int off = 16; off > 0; off >>= 1) {
    const float mo = __shfl_xor(m, off, 32);
    const float so = __shfl_xor(s, off, 32);
    const float M  = fmaxf(m, mo);
    s = s * __expf(m - M) + so * __expf(mo - M);
    m = M;
  }
  if (lane == 0) { Mrow[row] = m; Lrow[row] = s; }
}

// ---------------------------------------------------------------------------
// Probability pair in f32 16x16x4 A-operand layout for one 16-row tile.
// ---------------------------------------------------------------------------
__device__ __forceinline__ v2f make_p(float f1i, float mi,
                                      const int* __restrict__ ar, int jx,
                                      v2f f2p)
{
  float e0 = f1i + f2p.x; e0 = (e0 >= 0.f) ? e0 : kAlpha * e0;
  float e1 = f1i + f2p.y; e1 = (e1 >= 0.f) ? e1 : kAlpha * e1;
  const int2 ad = *(const int2*)(ar + jx);
  e0 = (ad.x > 0) ? e0 : kNegInf;
  e1 = (ad.y > 0) ? e1 : kNegInf;
  v2f p;
  p.x = __expf(e0 - mi);                           // exp(NEG_INF - m) -> 0
  p.y = __expf(e1 - mi);
  return p;
}

// ---------------------------------------------------------------------------
// Kernel 4: h' = softmax(e) @ H, out = elu(h').
// One wave per 32-row tile: two 16-row A (probability) tiles share the
// B-operand (h) loads.  K-loop over j: 1024/4 = 256 steps x 8 WMMAs.
// ---------------------------------------------------------------------------
__global__ __launch_bounds__(256) void k4_attn_av(
    const float* __restrict__ H, const float* __restrict__ F1,
    const float* __restrict__ F2, const int* __restrict__ adj,
    const float* __restrict__ abp, const float* __restrict__ Mrow,
    const float* __restrict__ Lrow, float* __restrict__ out)
{
  const int lane = threadIdx.x & 31;
  const int wave = threadIdx.x >> 5;
  const int gt   = blockIdx.x * 8 + wave;          // 0..255
  const int b    = gt >> 5;                        // 32 tiles per batch
  const int i0   = (gt & 31) << 5;                 // 32 rows per wave
  const int half = lane >> 4;
  const int ncol = lane & 15;
  const int irow0 = (b << 10) + i0 + ncol;         // A-tile 0 row for this lane
  const int irow1 = irow0 + 16;                    // A-tile 1 row
  const float ab   = abp[0];
  const float f1a  = F1[irow0] + ab;
  const float f1b  = F1[irow1] + ab;
  const float mia  = Mrow[irow0];
  const float mib  = Mrow[irow1];
  const int*   __restrict__ ar0 = adj + (size_t)irow0 * kN;
  const int*   __restrict__ ar1 = adj + (size_t)irow1 * kN;
  const float* __restrict__ f2b = F2 + (b << 10);
  const float* __restrict__ hb  = H + ((size_t)(b << 10)) * kOutF;

  v8f acc0[4] = {};
  v8f acc1[4] = {};
  for (int j0 = 0; j0 < kN; j0 += 4) {
    const int jx = j0 + half * 2;
    const v2f f2p = *(const v2f*)(f2b + jx);
    const v2f p0 = make_p(f1a, mia, ar0, jx, f2p);
    const v2f p1 = make_p(f1b, mib, ar1, jx, f2p);
    const float* __restrict__ hj = hb + (size_t)jx * kOutF;
#pragma unroll
    for (int t = 0; t < 4; ++t) {
      v2f bm;
      bm.x = hj[t * 16 + ncol];
      bm.y = hj[kOutF + t * 16 + ncol];
      acc0[t] = WMMA_F32_4(p0, bm, acc0[t]);
      acc1[t] = WMMA_F32_4(p1, bm, acc1[t]);
    }
  }
  const int rbase = (b << 10) + i0 + half * 8;
#pragma unroll
  for (int r = 0; r < 8; ++r) {
    const float li0 = 1.0f / Lrow[rbase + r];
    const float li1 = 1.0f / Lrow[rbase + 16 + r];
#pragma unroll
    for (int t = 0; t < 4; ++t) {
      float v0 = acc0[t][r] * li0;                 // softmax normalize
      float v1 = acc1[t][r] * li1;
      v0 = (v0 > 0.f) ? v0 : (__expf(v0) - 1.f);   // elu
      v1 = (v1 > 0.f) ? v1 : (__expf(v1) - 1.f);
      out[(size_t)(rbase + r) * kOutF + t * 16 + ncol]      = v0;
      out[(size_t)(rbase + 16 + r) * kOutF + t * 16 + ncol] = v1;
    }
  }
}

// ---------------------------------------------------------------------------
extern "C" void kernel_launch(void* const* d_in, const int* in_sizes, int n_in,
                              void* d_out, int out_size, void* d_ws, size_t ws_size,
                              hipStream_t stream) {
  (void)in_sizes; (void)n_in; (void)out_size; (void)ws_size;
  const float* X   = (const float*)d_in[0];  // (8,1024,768)
  // d_in[1]=tag, d_in[2]=attention_mask: unused by the reference
  const int*   adj = (const int*)d_in[3];    // (8,1024,1024)
  const float* Ww  = (const float*)d_in[4];  // (64,768)
  const float* Wb  = (const float*)d_in[5];  // (64)
  const float* aw  = (const float*)d_in[6];  // (1,128)
  const float* ab  = (const float*)d_in[7];  // (1)

  float* ws = (float*)d_ws;
  float* H  = ws;                                     // 8192*64
  float* F1 = H  + (size_t)kB * kN * kOutF;           // 8192
  float* F2 = F1 + (size_t)kB * kN;                   // 8192
  float* Mr = F2 + (size_t)kB * kN;                   // 8192
  float* Lr = Mr + (size_t)kB * kN;                   // 8192
  float* O  = (float*)d_out;

  hipLaunchKernelGGL(k1_gemm_h,  dim3(32),   dim3(256), 0, stream, X, Ww, Wb, H);
  hipLaunchKernelGGL(k2_f12,     dim3(32),   dim3(256), 0, stream, H, aw, F1, F2);
  hipLaunchKernelGGL(k3_stats,   dim3(1024), dim3(256), 0, stream, F1, F2, adj, ab, Mr, Lr);
  hipLaunchKernelGGL(k4_attn_av, dim3(32),   dim3(256), 0, stream, H, F1, F2, adj, ab, Mr, Lr, O);
}